// TestAllReduceRMSNormModel_7095285973068
// MI455X (gfx1250) — compile-verified
//
#include <hip/hip_runtime.h>
#include <cstdint>

#define TP       4
#define TOKENS   8192
#define HIDDEN   4096
#define EPS_RMS  1e-6f
#define FP8MAX   448.0f

#define T1 256                    // pass-1 threads (8 waves of 32)
#define C1 (HIDDEN / (T1 * 4))    // = 4 float4 chunks per thread
#define T2 256
#define C2 4

#if defined(__HIP_DEVICE_COMPILE__) && defined(__gfx1250__)
#define GFX1250_DEV 1
#else
#define GFX1250_DEV 0
#endif

#if GFX1250_DEV && __has_builtin(__builtin_amdgcn_global_load_async_to_lds_b128)
#define HAVE_ASYNC_BUILTIN 1
#else
#define HAVE_ASYNC_BUILTIN 0
#endif

typedef int   v4i __attribute__((vector_size(4 * sizeof(int))));
typedef float v4f __attribute__((ext_vector_type(4)));

// gfx12-style CPol: TH in bits [2:0]; TH=1 -> non-temporal (don't pollute L2
// with the 512MB one-shot input stream; we want the 128MB intermediate to
// stay resident in the 192MB L2 for pass 2).
#define CPOL_TH_NT 1

// ---- CDNA5 async global -> LDS copy (128-bit per lane), tracked by ASYNCcnt ----
__device__ __forceinline__ void async_copy_b128_nt(const void* gsrc, void* lds_dst) {
#if HAVE_ASYNC_BUILTIN
    __builtin_amdgcn_global_load_async_to_lds_b128(
        (__attribute__((address_space(1))) v4i*)gsrc,
        (__attribute__((address_space(3))) v4i*)lds_dst,
        /*offset=*/0, /*cpol=*/CPOL_TH_NT);
#elif GFX1250_DEV
    unsigned           laddr = (unsigned)(uintptr_t)lds_dst;        // low 32 bits = LDS offset
    unsigned long long gaddr = (unsigned long long)(uintptr_t)gsrc;
    asm volatile("global_load_async_to_lds_b128 %0, %1, off th:TH_LOAD_NT"
                 :: "v"(laddr), "v"(gaddr) : "memory");
#else
    *(float4*)lds_dst = *(const float4*)gsrc;  // host-pass placeholder only
#endif
}

__device__ __forceinline__ void async_wait0() {
#if GFX1250_DEV
    asm volatile("s_wait_asynccnt 0x0" ::: "memory");
#endif
}

// ---- kernel 0: zero the amax accumulator in workspace ----
__global__ void ar_rms_init_kernel(unsigned* amax_bits) {
    if (threadIdx.x == 0 && blockIdx.x == 0) amax_bits[0] = 0u;
}

// ---- kernel 1: TP-sum + RMSNorm + per-block amax; one block per token row ----
__global__ __launch_bounds__(T1) void ar_rms_pass1_kernel(
        const float* __restrict__ hs,     // [TP][TOKENS][HIDDEN]
        const float* __restrict__ w,      // [HIDDEN]
        float* __restrict__ out,          // [TOKENS][HIDDEN] (+1 scale slot)
        unsigned* __restrict__ amax_bits) {
    __shared__ float4 stage[TP][T1 * C1];   // 4 * 16KB = 64KB staging
    __shared__ float  red[T1 / 32];
    __shared__ float  s_inv;

    const int t    = threadIdx.x;
    const int row  = blockIdx.x;
    const int lane = t & 31;
    const int wave = t >> 5;

    const float4* hs4  = (const float4*)hs;
    const long long row4 = (long long)row * (HIDDEN / 4);
    const long long sh4  = (long long)TOKENS * (HIDDEN / 4);

    // Issue all async shard-row copies up front (16 in-flight b128 ops/thread).
    #pragma unroll
    for (int c = 0; c < C1; ++c) {
        const int idx = c * T1 + t;
        #pragma unroll
        for (int s = 0; s < TP; ++s)
            async_copy_b128_nt(&hs4[(long long)s * sh4 + row4 + idx], &stage[s][idx]);
    }
    async_wait0();   // each thread consumes only the LDS it wrote -> no barrier needed

    // TP all-reduce into registers + sum of squares.
    float4 v[C1];
    float sumsq = 0.f;
    #pragma unroll
    for (int c = 0; c < C1; ++c) {
        const int idx = c * T1 + t;
        const float4 a = stage[0][idx];
        const float4 b = stage[1][idx];
        const float4 g = stage[2][idx];
        const float4 d = stage[3][idx];
        float4 sv;
        sv.x = a.x + b.x + g.x + d.x;
        sv.y = a.y + b.y + g.y + d.y;
        sv.z = a.z + b.z + g.z + d.z;
        sv.w = a.w + b.w + g.w + d.w;
        v[c] = sv;
        sumsq = fmaf(sv.x, sv.x, sumsq);
        sumsq = fmaf(sv.y, sv.y, sumsq);
        sumsq = fmaf(sv.z, sv.z, sumsq);
        sumsq = fmaf(sv.w, sv.w, sumsq);
    }

    // Block reduction of sum of squares (wave32 shuffle, then 8-wave LDS).
    #pragma unroll
    for (int off = 16; off > 0; off >>= 1) sumsq += __shfl_xor(sumsq, off, 32);
    if (lane == 0) red[wave] = sumsq;
    __syncthreads();
    if (t == 0) {
        float s = 0.f;
        #pragma unroll
        for (int i = 0; i < T1 / 32; ++i) s += red[i];
        s_inv = rsqrtf(s * (1.0f / (float)HIDDEN) + EPS_RMS);
    }
    __syncthreads();
    const float inv = s_inv;

    // Normalize, apply weight, write fp32 (default RT -> stays in L2), track |max|.
    const float4* w4   = (const float4*)w;
    float4*       out4 = (float4*)out;
    float am = 0.f;
    #pragma unroll
    for (int c = 0; c < C1; ++c) {
        const int idx = c * T1 + t;
        const float4 wv = w4[idx];
        float4 o;
        o.x = v[c].x * inv * wv.x;
        o.y = v[c].y * inv * wv.y;
        o.z = v[c].z * inv * wv.z;
        o.w = v[c].w * inv * wv.w;
        am = fmaxf(am, fmaxf(fmaxf(fabsf(o.x), fabsf(o.y)),
                             fmaxf(fabsf(o.z), fabsf(o.w))));
        out4[row4 + idx] = o;
    }
    #pragma unroll
    for (int off = 16; off > 0; off >>= 1) am = fmaxf(am, __shfl_xor(am, off, 32));
    if (lane == 0) red[wave] = am;
    __syncthreads();
    if (t == 0) {
        float m = red[0];
        #pragma unroll
        for (int i = 1; i < T1 / 32; ++i) m = fmaxf(m, red[i]);
        // all values >= 0 -> uint bit pattern is order-preserving
        atomicMax(amax_bits, __float_as_uint(m));
    }
}

// ---- kernel 2: global scale + clamp, in place on d_out; emit scale scalar ----
__global__ __launch_bounds__(T2) void ar_rms_pass2_kernel(
        float* __restrict__ out, const unsigned* __restrict__ amax_bits) {
    const float amax  = __uint_as_float(amax_bits[0]);
    const float scale = fmaxf(amax, 1e-12f) / FP8MAX;
    const float rs    = 1.0f / scale;

    v4f* out4 = (v4f*)out;
    const long long base = (long long)blockIdx.x * (T2 * C2);
    #pragma unroll
    for (int c = 0; c < C2; ++c) {
        const long long idx = base + c * T2 + threadIdx.x;
        v4f x = out4[idx];                 // hopefully L2-resident from pass 1
        v4f q;
        q.x = fminf(fmaxf(x.x * rs, -FP8MAX), FP8MAX);
        q.y = fminf(fmaxf(x.y * rs, -FP8MAX), FP8MAX);
        q.z = fminf(fmaxf(x.z * rs, -FP8MAX), FP8MAX);
        q.w = fminf(fmaxf(x.w * rs, -FP8MAX), FP8MAX);
        __builtin_nontemporal_store(q, &out4[idx]);   // final result, never re-read
    }
    if (blockIdx.x == 0 && threadIdx.x == 0)
        out[(long long)TOKENS * HIDDEN] = scale;
}

extern "C" void kernel_launch(void* const* d_in, const int* in_sizes, int n_in,
                              void* d_out, int out_size, void* d_ws, size_t ws_size,
                              hipStream_t stream) {
    const float* hs = (const float*)d_in[0];   // [TP][TOKENS][HIDDEN]
    // d_in[1] = residual: unused by the reference computation
    const float* w  = (const float*)d_in[2];   // [HIDDEN]
    float* out = (float*)d_out;                // TOKENS*HIDDEN values + 1 scale
    unsigned* amax_bits = (unsigned*)d_ws;

    ar_rms_init_kernel<<<1, 1, 0, stream>>>(amax_bits);
    ar_rms_pass1_kernel<<<TOKENS, T1, 0, stream>>>(hs, w, out, amax_bits);

    const int n4      = (TOKENS * HIDDEN) / 4;       // 8,388,608 float4
    const int blocks2 = n4 / (T2 * C2);              // 8192
    ar_rms_pass2_kernel<<<blocks2, T2, 0, stream>>>(out, amax_bits);
}